// GlobalLAPool_56435870270132
// MI455X (gfx1250) — compile-verified
//
#include <hip/hip_runtime.h>
#include <math.h>

// Problem constants (match reference)
#define NUM_GRAPHS 16384
#define C 128
#define TWO_C 256

// Pass-1 tiling
#define TILE_ROWS 64
#define LDS_STRIDE 132   // floats: 528B row stride -> 16B aligned, spreads LDS banks

typedef __attribute__((ext_vector_type(2))) float v2f;
typedef __attribute__((ext_vector_type(4))) float v4f;
typedef __attribute__((ext_vector_type(8))) float v8f;

__device__ __forceinline__ float wave_max(float v) {
#pragma unroll
    for (int off = 16; off > 0; off >>= 1) v = fmaxf(v, __shfl_xor(v, off, 32));
    return v;
}
__device__ __forceinline__ float wave_sum(float v) {
#pragma unroll
    for (int off = 16; off > 0; off >>= 1) v += __shfl_xor(v, off, 32);
    return v;
}

// ---------------------------------------------------------------------------
// Pass 1: per-graph online segment-softmax pooling.
// One 128-thread block (4 wave32) per graph. batch is sorted, so each graph's
// rows are a contiguous slice of x; we find it by binary search, stream it
// through LDS once, and maintain a running (m, d, acc) online softmax so any
// segment length is handled with a single read of x (memory-optimal:
// 512 MB @ 23.3 TB/s ~ 22 us dominates total runtime).
// Output: xbar[g][c] = sum_i softmax(gate)_i * x_i[c], sflag[g] = segment nonempty.
// ---------------------------------------------------------------------------
__global__ __launch_bounds__(128)
void seg_softmax_pool_kernel(const float* __restrict__ x,
                             const int*   __restrict__ batch,
                             const float* __restrict__ gate_w,
                             int N,
                             float* __restrict__ xbar,
                             float* __restrict__ sflag) {
    __shared__ float lds_x[TILE_ROWS * LDS_STRIDE];
    __shared__ float lds_gate[TILE_ROWS];
    __shared__ float lds_w[TILE_ROWS];
    __shared__ float lds_gw[C];
    __shared__ float red[8];

    const int g    = blockIdx.x;
    const int tid  = threadIdx.x;
    const int lane = tid & 31;
    const int wave = tid >> 5;

    lds_gw[tid] = gate_w[tid];

    // lower_bound(batch, g) and lower_bound(batch, g+1); uniform address
    // loads -> served by scalar/L1 path, done redundantly by all threads.
    int lo = 0, hi = N;
    while (lo < hi) { int mid = (lo + hi) >> 1; if (batch[mid] <  g) lo = mid + 1; else hi = mid; }
    const int start = lo;
    hi = N;
    while (lo < hi) { int mid = (lo + hi) >> 1; if (batch[mid] <= g) lo = mid + 1; else hi = mid; }
    const int end = lo;

    float run_m = -INFINITY;   // running max (replicated across threads)
    float run_d = 0.0f;        // running softmax denominator (replicated)
    float acc   = 0.0f;        // this thread's channel accumulator

    for (int t0 = start; t0 < end; t0 += TILE_ROWS) {
        const int rcount = min(TILE_ROWS, end - t0);

        __syncthreads();   // prev-tile readers of lds_x / lds_w done

        // Cooperative float4 load: segment rows are contiguous in memory.
        const v4f* src = (const v4f*)(x + (size_t)t0 * C);
        const int n4 = rcount * (C / 4);
        for (int j = tid; j < n4; j += 128) {
            const int r  = j >> 5;       // 32 float4 per row
            const int c4 = j & 31;
            *(v4f*)&lds_x[r * LDS_STRIDE + c4 * 4] = src[j];
        }
        __syncthreads();

        // Gate scores: one row per wave, lane-parallel 128-elem dot from LDS.
        for (int r = wave; r < rcount; r += 4) {
            const float* row = &lds_x[r * LDS_STRIDE];
            float p = fmaf(row[lane],      lds_gw[lane],
                      fmaf(row[lane + 32], lds_gw[lane + 32],
                      fmaf(row[lane + 64], lds_gw[lane + 64],
                           row[lane + 96] * lds_gw[lane + 96])));
            p = wave_sum(p);
            if (lane == 0) lds_gate[r] = p;
        }
        __syncthreads();

        // Tile max (rcount <= 64 <= 128 threads).
        const float gv = (tid < rcount) ? lds_gate[tid] : -INFINITY;
        const float wm = wave_max(gv);
        if (lane == 0) red[wave] = wm;
        __syncthreads();

        const float tile_m = fmaxf(fmaxf(red[0], red[1]), fmaxf(red[2], red[3]));
        const float new_m  = fmaxf(run_m, tile_m);
        const float scale  = __expf(run_m - new_m);   // 0 when run_m == -inf

        float ev = 0.0f;
        if (tid < rcount) { ev = __expf(gv - new_m); lds_w[tid] = ev; }
        const float wsum = wave_sum(ev);
        if (lane == 0) red[4 + wave] = wsum;
        __syncthreads();

        const float tile_d = red[4] + red[5] + red[6] + red[7];
        run_d = run_d * scale + tile_d;
        run_m = new_m;

        // Weighted accumulate: thread tid owns channel tid.
        acc *= scale;
        for (int r = 0; r < rcount; ++r)
            acc = fmaf(lds_w[r], lds_x[r * LDS_STRIDE + tid], acc);
    }

    const bool nonempty = (end > start);
    const float inv = nonempty ? (1.0f / run_d) : 0.0f;
    xbar[(size_t)g * C + tid] = acc * inv;
    if (tid == 0) sflag[g] = nonempty ? 1.0f : 0.0f;
}

// ---------------------------------------------------------------------------
// Pass 2: out = xbar @ nn_w + sflag * nn_b  via V_WMMA_F32_16X16X4_F32.
// [16384 x 128] x [128 x 256] -> [16384 x 256]. One 16x16 D tile per wave,
// 32 K-steps of k=4. xbar (8 MB) + nn_w (128 KB) live in the 192 MB L2.
// Fragment layouts per ISA 7.12.2:
//   A (16x4 f32):  element (m,k) -> lane = m + 16*(k>>1), vgpr = k&1
//   B (4x16 f32):  element (k,n) -> lane = n + 16*(k>>1), vgpr = k&1
//   C/D (16x16):   element (m,n) -> vgpr = m%8, lane = n + 16*(m/8)
// ---------------------------------------------------------------------------
__global__ __launch_bounds__(256)
void wmma_out_gemm_kernel(const float* __restrict__ xbar,
                          const float* __restrict__ nn_w,
                          const float* __restrict__ nn_b,
                          const float* __restrict__ sflag,
                          float* __restrict__ out) {
    const int tid  = threadIdx.x;
    const int lane = tid & 31;
    const int wave = tid >> 5;

    const int tiles_n = TWO_C / 16;                 // 16
    const int tile = blockIdx.x * 8 + wave;         // 16384 tiles total
    const int rb = (tile / tiles_n) * 16;           // output row base
    const int cb = (tile % tiles_n) * 16;           // output col base

    const int mn   = lane & 15;                     // A-row / B-col index
    const int ksel = lane >> 4;                     // which K pair this half-wave holds

    v8f c = {};
#pragma unroll 4
    for (int kb = 0; kb < C; kb += 4) {
        const float* ap = xbar + (size_t)(rb + mn) * C + kb + 2 * ksel;
        v2f a; a.x = ap[0]; a.y = ap[1];
        const float* bp = nn_w + (size_t)(kb + 2 * ksel) * TWO_C + cb + mn;
        v2f b; b.x = bp[0]; b.y = bp[TWO_C];
        c = __builtin_amdgcn_wmma_f32_16x16x4_f32(false, a, false, b,
                                                  (short)0, c, false, false);
    }

    const float bias = nn_b[cb + mn];
    const int mofs = 8 * ksel;
#pragma unroll
    for (int v = 0; v < 8; ++v) {
        const int row = rb + v + mofs;
        out[(size_t)row * TWO_C + cb + mn] = c[v] + sflag[row] * bias;
    }
}

// ---------------------------------------------------------------------------
extern "C" void kernel_launch(void* const* d_in, const int* in_sizes, int n_in,
                              void* d_out, int out_size, void* d_ws, size_t ws_size,
                              hipStream_t stream) {
    (void)n_in; (void)out_size; (void)ws_size;
    const float* x      = (const float*)d_in[0];
    const int*   batch  = (const int*)  d_in[1];
    const float* gate_w = (const float*)d_in[2];
    // d_in[3] = gate_b: softmax is shift-invariant -> unused
    const float* nn_w   = (const float*)d_in[4];
    const float* nn_b   = (const float*)d_in[5];
    float* out = (float*)d_out;

    const int N = in_sizes[0] / C;

    float* xbar  = (float*)d_ws;                         // [NUM_GRAPHS, C]
    float* sflag = xbar + (size_t)NUM_GRAPHS * C;        // [NUM_GRAPHS]

    seg_softmax_pool_kernel<<<NUM_GRAPHS, 128, 0, stream>>>(
        x, batch, gate_w, N, xbar, sflag);

    const int total_tiles = (NUM_GRAPHS / 16) * (TWO_C / 16);   // 16384
    wmma_out_gemm_kernel<<<total_tiles / 8, 256, 0, stream>>>(
        xbar, nn_w, nn_b, sflag, out);
}